// GraphConvLayer_9586367005311
// MI455X (gfx1250) — compile-verified
//
#include <hip/hip_runtime.h>
#include <hip/hip_bf16.h>

typedef float v2f __attribute__((ext_vector_type(2)));
typedef float v8f __attribute__((ext_vector_type(8)));

#define DIM 128
#define NNODES 100000

// ---------------------------------------------------------------------------
// Kernel 1: zero the aggregation buffer (workspace is poisoned, must clear).
// ---------------------------------------------------------------------------
__global__ __launch_bounds__(256) void gc_zero_kernel(float4* __restrict__ p, int n4) {
    int i = blockIdx.x * blockDim.x + threadIdx.x;
    if (i < n4) p[i] = make_float4(0.f, 0.f, 0.f, 0.f);
}

// ---------------------------------------------------------------------------
// Kernel 2: scatter-add  agg[dst[e]] += x[src[e]]
// One 32-lane group per edge; each lane handles 4 consecutive features
// (float4 coalesced gather, 4x global_atomic_add_f32 scatter).
// ---------------------------------------------------------------------------
__global__ __launch_bounds__(256) void gc_scatter_kernel(
        const float* __restrict__ x,
        const int* __restrict__ src,
        const int* __restrict__ dst,
        float* __restrict__ agg,
        int E) {
    int t = blockIdx.x * blockDim.x + threadIdx.x;
    int e = t >> 5;
    if (e >= E) return;
    int f = (t & 31) * 4;
    int s = src[e];
    int d = dst[e];
    const float4 v = *(const float4*)(x + (size_t)s * DIM + f);
    float* o = agg + (size_t)d * DIM + f;
    atomicAdd(o + 0, v.x);
    atomicAdd(o + 1, v.y);
    atomicAdd(o + 2, v.z);
    atomicAdd(o + 3, v.w);
}

// ---------------------------------------------------------------------------
// Kernel 3: out = agg @ W_rel^T + x @ W_root^T + b_rel   (full fp32 via WMMA)
//
// Block = 256 threads = 8 waves. Block owns 16 rows (one M tile); each wave
// owns one 16-wide N tile. A tiles (agg, x) staged once into LDS; B read from
// global (weights are 128KB, L2 resident). Two independent WMMA accumulation
// chains per wave for XDL pipelining.
//
// V_WMMA_F32_16X16X4_F32 operand layout (ISA 7.12.2):
//   A (16x4):  lane L, vgpr v -> (M = L%16, K = 2*(L/16) + v)
//   B (4x16):  lane L, vgpr v -> (K = 2*(L/16) + v, N = L%16)
//   C/D:       lane L, vgpr v -> (M = v + 8*(L/16), N = L%16)
// out = A @ W^T  =>  B[k][n] = W[n][k], W row-major [out_dim][in_dim].
// ---------------------------------------------------------------------------
__global__ __launch_bounds__(256) void gc_gemm_kernel(
        const float* __restrict__ agg,
        const float* __restrict__ x,
        const float* __restrict__ Wrel,
        const float* __restrict__ Wroot,
        const float* __restrict__ brel,
        float* __restrict__ out) {
    __shared__ __align__(16) float lds_agg[16 * DIM];
    __shared__ __align__(16) float lds_x[16 * DIM];

    const int m0  = blockIdx.x * 16;
    const int tid = threadIdx.x;

    // Cooperative stage of the 16x128 A tiles: 512 float4 per matrix,
    // 2 float4 per thread per matrix.
    {
        const float4* aggG = (const float4*)(agg + (size_t)m0 * DIM);
        const float4* xG   = (const float4*)(x   + (size_t)m0 * DIM);
        float4* aggL = (float4*)lds_agg;
        float4* xL   = (float4*)lds_x;
#pragma unroll
        for (int i = 0; i < 2; ++i) {
            int idx = tid + i * 256;      // 0..511
            aggL[idx] = aggG[idx];
            xL[idx]   = xG[idx];
        }
    }
    __syncthreads();

    const int wave = tid >> 5;            // N tile id, 0..7
    const int lane = tid & 31;
    const int half = lane >> 4;           // 0 or 1
    const int l16  = lane & 15;
    const int n0   = wave * 16;
    const int kb   = half * 2;            // lane's K sub-offset within each K=4 step

    const float* aRowAgg  = lds_agg + l16 * DIM;
    const float* aRowX    = lds_x   + l16 * DIM;
    const float* bRowRel  = Wrel  + (size_t)(n0 + l16) * DIM;
    const float* bRowRoot = Wroot + (size_t)(n0 + l16) * DIM;

    v8f accR = {};   // agg @ W_rel^T chain
    v8f accX = {};   // x   @ W_root^T chain

#pragma unroll
    for (int kk = 0; kk < DIM; kk += 4) {
        v2f aR, aX, bR, bX;
        aR.x = aRowAgg[kk + kb];     aR.y = aRowAgg[kk + kb + 1];
        bR.x = bRowRel[kk + kb];     bR.y = bRowRel[kk + kb + 1];
        aX.x = aRowX[kk + kb];       aX.y = aRowX[kk + kb + 1];
        bX.x = bRowRoot[kk + kb];    bX.y = bRowRoot[kk + kb + 1];
        accR = __builtin_amdgcn_wmma_f32_16x16x4_f32(
            false, aR, false, bR, (short)0, accR, false, false);
        accX = __builtin_amdgcn_wmma_f32_16x16x4_f32(
            false, aX, false, bX, (short)0, accX, false, false);
    }

    const float bias = brel[n0 + l16];

    // C/D layout: vgpr v holds (M = v + 8*half, N = l16)
#pragma unroll
    for (int v = 0; v < 8; ++v) {
        int m = m0 + v + 8 * half;
        out[(size_t)m * DIM + n0 + l16] = accR[v] + accX[v] + bias;
    }
}

// ---------------------------------------------------------------------------
// Harness entry point.
// Inputs (setup_inputs order): x[100000,128] f32, edge_index[2,600000] int,
// W_rel[128,128] f32, W_root[128,128] f32, b_rel[128] f32.
// ---------------------------------------------------------------------------
extern "C" void kernel_launch(void* const* d_in, const int* in_sizes, int n_in,
                              void* d_out, int out_size, void* d_ws, size_t ws_size,
                              hipStream_t stream) {
    const float* x     = (const float*)d_in[0];
    const int*   ei    = (const int*)d_in[1];
    const float* Wrel  = (const float*)d_in[2];
    const float* Wroot = (const float*)d_in[3];
    const float* brel  = (const float*)d_in[4];
    float*       out   = (float*)d_out;
    float*       agg   = (float*)d_ws;          // 100000*128*4 = 51.2 MB scratch

    const int E   = in_sizes[1] / 2;            // 600000
    const int* src = ei;
    const int* dst = ei + E;

    // 1) agg = 0
    const int n4 = NNODES * DIM / 4;            // 3.2M float4
    gc_zero_kernel<<<(n4 + 255) / 256, 256, 0, stream>>>((float4*)agg, n4);

    // 2) agg[dst] += x[src]   (32 lanes per edge)
    const int scatterThreads = E * 32;
    gc_scatter_kernel<<<(scatterThreads + 255) / 256, 256, 0, stream>>>(
        x, src, dst, agg, E);

    // 3) out = agg @ Wrel^T + x @ Wroot^T + brel   (100000/16 = 6250 M tiles)
    gc_gemm_kernel<<<NNODES / 16, 256, 0, stream>>>(agg, x, Wrel, Wroot, brel, out);
}